// SpecDCT_29523605192890
// MI455X (gfx1250) — compile-verified
//
#include <hip/hip_runtime.h>
#include <hip/hip_bf16.h>
#include <math.h>
#include <stdint.h>

// ---------------------------------------------------------------------------
// SpecDCT pipeline for MI455X (gfx1250):
//  - all matrix math on v_wmma_f32_16x16x4_f32 (exact f32, matches reference)
//  - GEMM B-tiles DMA'd into LDS by the Tensor Data Mover, double-buffered so
//    tensor_load_to_lds overlaps the WMMA inner loop (s_wait_tensorcnt sync)
// ---------------------------------------------------------------------------

typedef __attribute__((ext_vector_type(2))) float v2f;
typedef __attribute__((ext_vector_type(8))) float v8f;
typedef __attribute__((ext_vector_type(4))) unsigned int v4u;
typedef __attribute__((ext_vector_type(4))) int v4i;
typedef __attribute__((ext_vector_type(8))) int v8i;

#define B_ 4
#define H_ 256
#define W_ 256
#define C_ 64
#define HW_ (H_ * W_)          // 65536
#define PLANE_ ((size_t)HW_)   // per-(b,c) plane, floats

#if defined(__gfx1250__) && __has_builtin(__builtin_amdgcn_tensor_load_to_lds) && \
    __has_builtin(__builtin_amdgcn_s_wait_tensorcnt)
#define SPEC_USE_TDM 1
#else
#define SPEC_USE_TDM 0
#endif

__device__ __forceinline__ float gelu_exact(float x) {
    return 0.5f * x * (1.0f + erff(x * 0.70710678118654752f));
}

__device__ __forceinline__ v8f wmma_f32(v2f a, v2f b, v8f c) {
    return __builtin_amdgcn_wmma_f32_16x16x4_f32(false, a, false, b, (short)0, c,
                                                 false, false);
}

#if SPEC_USE_TDM
// TDM: DMA a 16-row x 256-f32 tile (constant row stride) from global into LDS.
// D# per CDNA5 ISA ch.8: group0 = {count|flags, lds_addr, global_addr, type=2},
// group1 = {data_size=4B, tensor dims == tile dims (no OOB), tile 256x16,
// tensor_dim0_stride = row stride}. Groups 2/3 zero (2-D tensor).
__device__ __forceinline__ void tdm_load_tile_16x256(unsigned lds_addr,
                                                     const float* src,
                                                     unsigned stride_elems) {
    unsigned long long ga = (unsigned long long)(uintptr_t)src;
    v4u g0;
    g0[0] = 1u;                                               // count=1 (user)
    g0[1] = lds_addr;                                         // LDS byte addr
    g0[2] = (unsigned)ga;                                     // global[31:0]
    g0[3] = (unsigned)((ga >> 32) & 0x1FFFFFFu) | (2u << 30); // global[56:32]|type=2
    v8i g1;
    g1[0] = (int)(2u << 16);        // data_size=2 (4 bytes), no multicast/flags
    g1[1] = (int)(256u << 16);      // tensor_dim0[15:0]=256 (abar addr = 0)
    g1[2] = (int)(16u << 16);       // tensor_dim0[31:16]=0 | tensor_dim1[15:0]=16
    g1[3] = (int)(256u << 16);      // tensor_dim1[31:16]=0 | tile_dim0=256
    g1[4] = 16;                     // tile_dim1=16, tile_dim2=0
    g1[5] = (int)stride_elems;      // tensor_dim0_stride[31:0]
    g1[6] = 0;                      // stride0[47:32]=0, stride1[15:0]=0
    g1[7] = 0;
    v4i gz4 = {};
#if __clang_major__ >= 23
    v8i gz8 = {};
    __builtin_amdgcn_tensor_load_to_lds(g0, g1, gz4, gz4, gz8, 0);
#else
    __builtin_amdgcn_tensor_load_to_lds(g0, g1, gz4, gz4, 0);
#endif
}
#endif

// ---------------------------------------------------------------------------
// Kernel 0: build DCT-II / inverse-DCT matrices (256x256) + transposes in ws
// D[k][n]  = 2*cos(pi*(2n+1)k/512)            (torch_dct norm=None forward)
// ID[n][k] = w[k]*cos(pi*(2n+1)k/512)/256     (exact inverse, w[0]=0.5)
// ---------------------------------------------------------------------------
__global__ void build_mats(float* __restrict__ D, float* __restrict__ Dt,
                           float* __restrict__ ID, float* __restrict__ IDt) {
    int k = blockIdx.x;       // frequency index
    int n = threadIdx.x;      // spatial index
    float ang = 3.14159265358979323846f * (float)((2 * n + 1) * k) / 512.0f;
    float c = cosf(ang);
    float dv = 2.0f * c;
    D[k * 256 + n]  = dv;     // D[kfreq][n]
    Dt[n * 256 + k] = dv;     // Dt[n][kfreq]
    float wk = (k == 0) ? 0.5f : 1.0f;
    float iv = wk * c * (1.0f / 256.0f);
    ID[n * 256 + k]  = iv;    // ID[n][kfreq]
    IDt[k * 256 + n] = iv;    // IDt[kfreq][n]
}

// ---------------------------------------------------------------------------
// Kernel 1: generic 16x256-tile WMMA GEMM, B tile via double-buffered TDM.
//   mode 0 ROW_FWD : out[r,k] = sum_w x_nhwc[r,w] * Dt[w,k]      (r=(b,c,h))
//   mode 1 COL     : out[m,w] = sum_h  mat[m,h]   * in[h,w]      per (b,c)
//   mode 2 CONV1   : h1[c,p]  = gelu(sum_ci W[c,ci] * x[ci,p])   per b, K=64
//   mode 3 ROW_INV : out[r,n] = sum_w in[r,w] * IDt[w,n]
//   mode 4 COL_OUT : d_out[b,n,w,c] = sum_h ID[n,h] * in[h,w]    (NHWC scatter)
// Grid: 4096 WGs x 128 threads (4 waves, 4 N-tiles each).
// ---------------------------------------------------------------------------
__global__ __launch_bounds__(128) void gemm16(const float* __restrict__ in,
                                              const float* __restrict__ mat,
                                              float* __restrict__ outp,
                                              int mode) {
    __shared__ float As[16 * 17];        // A tile, padded stride 17 (bank-safe)
    __shared__ float Bs[2 * 16 * 256];   // double-buffered B tile (TDM dest)

    const int tid   = threadIdx.x;
    const int wave  = tid >> 5;
    const int lane  = tid & 31;
    const int lhalf = lane & 15;
    const int hi    = lane >> 4;      // 0: lanes 0-15, 1: lanes 16-31
    const int bid   = blockIdx.x;

    int kblocks = 16;
    int rbase = 0, mtile = 0, plane = 0, bidx = 0, chunk = 0;
    if (mode == 0 || mode == 3) {
        rbase = bid * 16;                       // 65536 rows / 16
    } else if (mode == 1 || mode == 4) {
        plane = bid >> 4;                       // (b*64 + c)
        mtile = bid & 15;
    } else {                                    // CONV1
        bidx  = bid >> 10;
        int r = bid & 1023;
        mtile = r >> 8;                         // 4 output-channel tiles
        chunk = r & 255;                        // 256-pixel chunk
        kblocks = 4;                            // K = 64
    }

    // B-tile source for K-block kb2 (uniform): pointer + row stride (elems)
    auto bsrc = [&](int kb2, unsigned& strideElems) -> const float* {
        if (mode == 0 || mode == 3) {
            strideElems = 256u;
            return mat + (size_t)kb2 * 16 * 256;
        } else if (mode == 1 || mode == 4) {
            strideElems = 256u;
            return in + (size_t)plane * PLANE_ + (size_t)kb2 * 16 * 256;
        }
        strideElems = 65536u;
        return in + ((size_t)bidx * C_ + (size_t)kb2 * 16) * PLANE_ +
               (size_t)chunk * 256;
    };

    v8f acc[4] = {};

#if SPEC_USE_TDM
    if (tid < 32) {                              // prologue: DMA tile 0
        unsigned se;
        const float* s = bsrc(0, se);
        tdm_load_tile_16x256((unsigned)(uintptr_t)Bs, s, se);
    }
#endif

    for (int kb = 0; kb < kblocks; ++kb) {
        __syncthreads();                // (1) WAR: prior-iter As/Bs reads done
        // ---- stage A tile (16 x 16) ----
        for (int idx = tid; idx < 256; idx += 128) {
            int rr = idx >> 4, kk = idx & 15;
            int kg = kb * 16 + kk;
            float v;
            if (mode == 0) {                    // gather from NHWC input
                int r = rbase + rr;
                int b = r >> 14, c = (r >> 8) & 63, h = r & 255;
                v = in[(((size_t)b * H_ + h) * W_ + kg) * C_ + c];
            } else if (mode == 3) {
                v = in[(size_t)(rbase + rr) * 256 + kg];
            } else if (mode == 1 || mode == 4) {
                v = mat[(mtile * 16 + rr) * 256 + kg];
            } else {                            // CONV1: 64x64 weight
                v = mat[(mtile * 16 + rr) * 64 + kg];
            }
            As[rr * 17 + kk] = v;
        }
#if SPEC_USE_TDM
        const float* Bcur = Bs + (size_t)(kb & 1) * 4096;
        if (tid < 32) __builtin_amdgcn_s_wait_tensorcnt((short)0); // tile kb in
        __syncthreads();                // (2) publish As + Bs[kb&1]
        if (tid < 32 && kb + 1 < kblocks) {      // overlap: DMA next tile
            unsigned se;
            const float* s = bsrc(kb + 1, se);
            tdm_load_tile_16x256(
                (unsigned)(uintptr_t)(Bs + (size_t)((kb + 1) & 1) * 4096), s, se);
        }
#else
        const float* Bcur = Bs;
        for (int idx = tid; idx < 4096; idx += 128) {
            int kk = idx >> 8, n = idx & 255;
            int kg = kb * 16 + kk;
            float v;
            if (mode == 0 || mode == 3) {
                v = mat[kg * 256 + n];
            } else if (mode == 1 || mode == 4) {
                v = in[(size_t)plane * PLANE_ + kg * 256 + n];
            } else {
                v = in[((size_t)bidx * C_ + kg) * PLANE_ + chunk * 256 + n];
            }
            Bs[kk * 256 + n] = v;
        }
        __syncthreads();                // publish As + Bs
#endif
        // ---- 4 K-steps of 4, 4 N-tiles per wave ----
#pragma unroll
        for (int ks = 0; ks < 4; ++ks) {
            int krow = ks * 4 + hi * 2;         // ISA A layout: K split by half
            v2f a;
            a.x = As[lhalf * 17 + krow];
            a.y = As[lhalf * 17 + krow + 1];
#pragma unroll
            for (int t = 0; t < 4; ++t) {
                int n0 = (wave * 4 + t) * 16;
                v2f bf;
                bf.x = Bcur[krow * 256 + n0 + lhalf];
                bf.y = Bcur[(krow + 1) * 256 + n0 + lhalf];
                acc[t] = wmma_f32(a, bf, acc[t]);
            }
        }
    }

    // ---- writeback (C layout: lane=N within tile, vgpr v -> row v + hi*8) ----
#pragma unroll
    for (int t = 0; t < 4; ++t) {
        int nc = (wave * 4 + t) * 16 + lhalf;
#pragma unroll
        for (int v = 0; v < 8; ++v) {
            int m = v + hi * 8;
            float val = acc[t][v];
            if (mode == 0 || mode == 3) {
                outp[(size_t)(rbase + m) * 256 + nc] = val;
            } else if (mode == 1) {
                outp[(size_t)plane * PLANE_ + (mtile * 16 + m) * 256 + nc] = val;
            } else if (mode == 4) {             // scatter to (b, n, w, c)
                int b = plane >> 6, c = plane & 63;
                outp[(((size_t)b * H_ + (mtile * 16 + m)) * W_ + nc) * C_ + c] = val;
            } else {                            // CONV1 + GELU
                outp[((size_t)bidx * C_ + (mtile * 16 + m)) * PLANE_ +
                     chunk * 256 + nc] = gelu_exact(val);
            }
        }
    }
}

// ---------------------------------------------------------------------------
// Kernel 2: per-8x8-block multi-head channel attention (one WG per block).
// 128 threads = 4 waves = 4 heads. All GEMMs on WMMA f32 16x16x4.
// ---------------------------------------------------------------------------
__global__ __launch_bounds__(128) void attn_kernel(
    const float* __restrict__ xdct, const float* __restrict__ Wq,
    const float* __restrict__ Wk, const float* __restrict__ Wv,
    const float* __restrict__ rescale, const float* __restrict__ projw,
    const float* __restrict__ projb, float* __restrict__ xlow) {
    __shared__ float xt[64 * 68];     // x_tri[m][c]; later reused as x1[m][hd]
    __shared__ float qs[64 * 68];     // q_t[m][hd]
    __shared__ float ksh[64 * 68];    // k_t[m][hd]
    __shared__ float vs[64 * 68];     // v_t[m][hd]
    __shared__ float at[4][16 * 16];  // per-head softmax(attn)

    const int bid = blockIdx.x;       // b*1024 + hb*32 + wb
    const int b  = bid >> 10;
    const int rr = bid & 1023;
    const int hb = rr >> 5;
    const int wb = rr & 31;
    const int tid   = threadIdx.x;
    const int wave  = tid >> 5;
    const int lane  = tid & 31;
    const int lhalf = lane & 15;
    const int hi    = lane >> 4;

    const float* pb = xdct + (size_t)b * C_ * PLANE_;

    // -- load x_tri: xt[m][c] = xdct[b, c, hb*8+i, wb*8+j], m = i*8+j
    for (int idx = tid; idx < 4096; idx += 128) {
        int c = idx >> 6, m = idx & 63;
        int i = m >> 3, j = m & 7;
        xt[m * 68 + c] = pb[(size_t)c * PLANE_ + (hb * 8 + i) * 256 + (wb * 8 + j)];
    }
    __syncthreads();

    // -- QKV projections: 3 matrices x (4x4) tiles = 48 tiles, 12 per wave
    for (int t = wave * 12; t < wave * 12 + 12; ++t) {
        int matid = t >> 4;
        int rem = t & 15;
        int mt = rem >> 2, nt = rem & 3;
        const float* W = (matid == 0) ? Wq : ((matid == 1) ? Wk : Wv);
        float* dst = (matid == 0) ? qs : ((matid == 1) ? ksh : vs);
        v8f acc = {};
#pragma unroll
        for (int ks = 0; ks < 16; ++ks) {
            int krow = ks * 4 + hi * 2;
            v2f a, bf;
            a.x  = xt[(mt * 16 + lhalf) * 68 + krow];
            a.y  = xt[(mt * 16 + lhalf) * 68 + krow + 1];
            bf.x = W[krow * 64 + nt * 16 + lhalf];
            bf.y = W[(krow + 1) * 64 + nt * 16 + lhalf];
            acc = wmma_f32(a, bf, acc);
        }
#pragma unroll
        for (int v = 0; v < 8; ++v)
            dst[(mt * 16 + v + hi * 8) * 68 + nt * 16 + lhalf] = acc[v];
    }
    __syncthreads();

    // -- L2-normalize q,k along m (columns of q_t/k_t); 128 cols, one/thread
    {
        int col = tid & 63;
        float* arr = (tid < 64) ? qs : ksh;
        float ss = 0.0f;
        for (int m = 0; m < 64; ++m) {
            float x = arr[m * 68 + col];
            ss += x * x;
        }
        float inv = 1.0f / fmaxf(sqrtf(ss), 1e-12f);
        for (int m = 0; m < 64; ++m) arr[m * 68 + col] *= inv;
    }
    __syncthreads();

    // -- per-head attention: wave = head
    const int h = wave;
    const float rs = rescale[h];
    v8f aacc = {};
#pragma unroll
    for (int ks = 0; ks < 16; ++ks) {
        int krow = ks * 4 + hi * 2;               // K dim = m
        v2f a, bf;
        a.x  = qs[krow * 68 + h * 16 + lhalf];    // A[d=lhalf, m]
        a.y  = qs[(krow + 1) * 68 + h * 16 + lhalf];
        bf.x = ksh[krow * 68 + h * 16 + lhalf];   // B[m, e=lhalf]
        bf.y = ksh[(krow + 1) * 68 + h * 16 + lhalf];
        aacc = wmma_f32(a, bf, aacc);
    }
    // softmax over e (16 lanes of each half hold one row's 16 e-values)
#pragma unroll
    for (int v = 0; v < 8; ++v) {
        float x = aacc[v] * rs;
        float mx = x;
        for (int o = 8; o >= 1; o >>= 1) mx = fmaxf(mx, __shfl_xor(mx, o, 16));
        float e = __expf(x - mx);
        float sm = e;
        for (int o = 8; o >= 1; o >>= 1) sm += __shfl_xor(sm, o, 16);
        at[h][(v + hi * 8) * 16 + lhalf] = e / sm;
    }
    __syncthreads();

    // -- out_h[d,m] = attn @ V  (M=16 d, N=64 m, K=16 e)
    v8f oacc[4] = {};
#pragma unroll
    for (int ks = 0; ks < 4; ++ks) {
        int krow = ks * 4 + hi * 2;
        v2f a;
        a.x = at[h][lhalf * 16 + krow];
        a.y = at[h][lhalf * 16 + krow + 1];
#pragma unroll
        for (int nt = 0; nt < 4; ++nt) {
            v2f bf;
            bf.x = vs[(nt * 16 + lhalf) * 68 + h * 16 + krow];
            bf.y = vs[(nt * 16 + lhalf) * 68 + h * 16 + krow + 1];
            oacc[nt] = wmma_f32(a, bf, oacc[nt]);
        }
    }
    __syncthreads();   // xt free: safe to reuse as x1
    // x1[m][h*16+d] (transpose of out_h)
#pragma unroll
    for (int nt = 0; nt < 4; ++nt)
#pragma unroll
        for (int v = 0; v < 8; ++v)
            xt[(nt * 16 + lhalf) * 68 + h * 16 + v + hi * 8] = oacc[nt][v];
    __syncthreads();

    // -- projection: x2[m,c] = x1 @ proj_w + b; wave handles m-tile = wave
    const int mt = wave;
#pragma unroll
    for (int nt = 0; nt < 4; ++nt) {
        v8f pacc = {};
#pragma unroll
        for (int ks = 0; ks < 16; ++ks) {
            int krow = ks * 4 + hi * 2;
            v2f a, bf;
            a.x  = xt[(mt * 16 + lhalf) * 68 + krow];
            a.y  = xt[(mt * 16 + lhalf) * 68 + krow + 1];
            bf.x = projw[krow * 64 + nt * 16 + lhalf];
            bf.y = projw[(krow + 1) * 64 + nt * 16 + lhalf];
            pacc = wmma_f32(a, bf, pacc);
        }
        int c = nt * 16 + lhalf;
        float bias = projb[c];
#pragma unroll
        for (int v = 0; v < 8; ++v) {
            int m = mt * 16 + v + hi * 8;
            int i = m >> 3, j = m & 7;
            xlow[((size_t)(b * C_ + c) * H_ + hb * 8 + i) * W_ + wb * 8 + j] =
                pacc[v] + bias;
        }
    }
}

// ---------------------------------------------------------------------------
// Kernel 3: depthwise 3x3 + GELU + residual (VALU), then 1x1 conv via WMMA,
// fused with x_high residual and coef mix -> x_mid. One WG per (b, h-row).
// ---------------------------------------------------------------------------
__global__ __launch_bounds__(256) void dwmix(
    const float* __restrict__ h1, const float* __restrict__ xdct,
    const float* __restrict__ xlow, const float* __restrict__ dww,
    const float* __restrict__ w2, const float* __restrict__ coef,
    float* __restrict__ outmid) {
    __shared__ float xc[64 * 256];     // x_conv slab (C x W), 64KB

    const int bid = blockIdx.x;
    const int b = bid >> 8;
    const int hrow = bid & 255;
    const int tid = threadIdx.x;
    const size_t base = (size_t)b * C_ * PLANE_;

    // phase 1: x_conv[c][w] = gelu(dw3x3(h1)) + x_dct
    for (int idx = tid; idx < 64 * 256; idx += 256) {
        int c = idx >> 8, w = idx & 255;
        const float* wp = dww + c * 9;
        float s = 0.0f;
#pragma unroll
        for (int dy = -1; dy <= 1; ++dy) {
            int hh = hrow + dy;
            if (hh < 0 || hh > 255) continue;
#pragma unroll
            for (int dx = -1; dx <= 1; ++dx) {
                int ww = w + dx;
                if (ww < 0 || ww > 255) continue;
                s += wp[(dy + 1) * 3 + (dx + 1)] *
                     h1[base + (size_t)c * PLANE_ + hh * 256 + ww];
            }
        }
        xc[idx] = gelu_exact(s) + xdct[base + (size_t)c * PLANE_ + hrow * 256 + w];
    }
    __syncthreads();

    // phase 2: h2 = gelu(W2 @ x_conv); x_high = h2 + x_conv; mix -> outmid
    const int wave = tid >> 5, lane = tid & 31;
    const int lhalf = lane & 15, hi = lane >> 4;
    for (int tt = 0; tt < 8; ++tt) {            // 64 tiles / 8 waves
        int t = wave * 8 + tt;
        int mt = t >> 4, nt = t & 15;
        v8f acc = {};
#pragma unroll
        for (int ks = 0; ks < 16; ++ks) {
            int krow = ks * 4 + hi * 2;
            v2f a, bf;
            a.x  = w2[(mt * 16 + lhalf) * 64 + krow];
            a.y  = w2[(mt * 16 + lhalf) * 64 + krow + 1];
            bf.x = xc[krow * 256 + nt * 16 + lhalf];
            bf.y = xc[(krow + 1) * 256 + nt * 16 + lhalf];
            acc = wmma_f32(a, bf, acc);
        }
        int w = nt * 16 + lhalf;
        float cf = coef[hrow * 256 + w];
#pragma unroll
        for (int v = 0; v < 8; ++v) {
            int c = mt * 16 + v + hi * 8;
            size_t off = base + (size_t)c * PLANE_ + hrow * 256 + w;
            float xcv = xc[c * 256 + w];
            float xhigh = gelu_exact(acc[v]) + xcv;
            outmid[off] = cf * xlow[off] + (1.0f - cf) * xhigh + xdct[off];
        }
    }
}

// ---------------------------------------------------------------------------
// Host-side launch orchestration (graph-capture safe: kernels only).
// ws layout (floats): D | Dt | ID | IDt | bufDCT | bufB | bufLOW
// ---------------------------------------------------------------------------
extern "C" void kernel_launch(void* const* d_in, const int* in_sizes, int n_in,
                              void* d_out, int out_size, void* d_ws,
                              size_t ws_size, hipStream_t stream) {
    const float* x_input = (const float*)d_in[0];
    const float* Wq      = (const float*)d_in[1];
    const float* Wk      = (const float*)d_in[2];
    const float* Wv      = (const float*)d_in[3];
    const float* rescale = (const float*)d_in[4];
    const float* proj_w  = (const float*)d_in[5];
    const float* proj_b  = (const float*)d_in[6];
    const float* c1a_w   = (const float*)d_in[7];
    const float* c1b_w   = (const float*)d_in[8];
    const float* c2_w    = (const float*)d_in[9];
    const float* coef    = (const float*)d_in[10];
    float* out = (float*)d_out;

    float* ws  = (float*)d_ws;
    float* D   = ws;
    float* Dt  = ws + 65536;
    float* ID  = ws + 131072;
    float* IDt = ws + 196608;
    float* bufDCT = ws + 262144;                    // (B,C,H,W) x_dct
    float* bufB   = bufDCT + (size_t)B_ * C_ * HW_; // scratch / h1
    float* bufLOW = bufB + (size_t)B_ * C_ * HW_;   // x_low -> x_mid

    build_mats<<<256, 256, 0, stream>>>(D, Dt, ID, IDt);
    // forward DCT: along W (rows, NHWC gather) then along H (per-plane)
    gemm16<<<4096, 128, 0, stream>>>(x_input, Dt, bufB, 0);
    gemm16<<<4096, 128, 0, stream>>>(bufB, D, bufDCT, 1);
    // block attention -> x_low
    attn_kernel<<<4096, 128, 0, stream>>>(bufDCT, Wq, Wk, Wv, rescale, proj_w,
                                          proj_b, bufLOW);
    // conv branch: 1x1+gelu -> h1 (bufB), then dw3x3 + 1x1 + mix -> x_mid
    gemm16<<<4096, 128, 0, stream>>>(bufDCT, c1a_w, bufB, 2);
    dwmix<<<1024, 256, 0, stream>>>(bufB, bufDCT, bufLOW, c1b_w, c2_w, coef,
                                    bufLOW);
    // inverse DCT: along W then along H with fused NHWC scatter to d_out
    gemm16<<<4096, 128, 0, stream>>>(bufLOW, IDt, bufB, 3);
    gemm16<<<4096, 128, 0, stream>>>(bufB, ID, out, 4);
}